// ParallelHyenaOperator_86706799772317
// MI455X (gfx1250) — compile-verified
//
#include <hip/hip_runtime.h>
#include <cstdint>

typedef __attribute__((ext_vector_type(2))) float v2f;
typedef __attribute__((ext_vector_type(8))) float v8f;
typedef __attribute__((ext_vector_type(4))) unsigned int v4u;
typedef __attribute__((ext_vector_type(8))) int v8i;
typedef __attribute__((ext_vector_type(4))) int v4i;

#define B_   2
#define D_   1024
#define L_   8192
#define G_   256
#define LC_  128
#define TT_  256   // time tile per workgroup
#define TC_  16    // channels per workgroup (4 groups)

// ---------------------------------------------------------------------------
// Kernel 1: decay filter precompute  k[g,tau] = h[g,tau]*exp(-10^(2g/255) * tau/127)
// ---------------------------------------------------------------------------
__global__ __launch_bounds__(LC_) void hyena_filter_precompute(
    const float* __restrict__ h, float* __restrict__ kbuf) {
  int g   = blockIdx.x;
  int tau = threadIdx.x;
  float e     = (2.0f * (float)g) * (1.0f / (float)(G_ - 1));
  float decay = __expf(e * 2.302585092994046f);           // 10^e
  float t     = (float)tau * (1.0f / (float)(LC_ - 1));
  kbuf[g * LC_ + tau] = h[g * LC_ + tau] * __expf(-decay * t);
}

// LDS skew: stride-16 accesses become stride-17 -> conflict-free on 64 banks
__device__ __forceinline__ int skew(int w) { return w + (w >> 4); }

// ---------------------------------------------------------------------------
// Kernel 2: fused causal 128-tap grouped FIR conv via V_WMMA_F32_16X16X4_F32
//           x1 tile staged by the Tensor Data Mover (pad == skew layout),
//           + bias shortcut + x1 gating + [B,D,L] -> [B,L,D] transpose store
// ---------------------------------------------------------------------------
__global__ __launch_bounds__(256) void hyena_conv_kernel(
    const float* __restrict__ x1, const float* __restrict__ x2,
    const float* __restrict__ v,  const float* __restrict__ kbuf,
    const float* __restrict__ conv_bias, float* __restrict__ out) {

  __shared__ float lds_x1[TC_ * 272];   // x1 window [T0, T0+256), TDM-padded (skewed)
  __shared__ float lds_k[4 * 160];      // zero-padded taps, tau in [-16,144) per group
  __shared__ float lds_z[TC_ * 408];    // z window [T0-128, T0+256), skewed
  __shared__ float lds_o[TT_ * TC_];    // output transpose staging (rotated rows)

  const int tid   = threadIdx.x;
  const int T0    = blockIdx.x * TT_;
  const int cbase = blockIdx.y * TC_;          // flattened b*D + d base
  const int b     = cbase >> 10;               // D_ = 1024
  const int dbase = cbase & (D_ - 1);
  const int g0    = dbase >> 2;                // 4 channels per group

  // ---- TDM: one DMA moves the whole 16x256 fp32 x1 tile into skewed LDS ----
  // pad_interval = 16 dwords, pad_amount = 1 dword  ==  skew(w) = w + (w>>4),
  // row stride 256+16 = 272 dwords, matching lds_x1 exactly.
  if (tid == 0) {
    uint64_t ga = (uint64_t)(uintptr_t)(x1 + (size_t)cbase * L_ + T0);
    uint32_t lb = (uint32_t)(uintptr_t)(&lds_x1[0]);
    v4u g0d;
    g0d[0] = 1u;                                    // count=1 valid descriptor
    g0d[1] = lb;                                    // lds_addr
    g0d[2] = (uint32_t)ga;                          // global_addr[31:0]
    g0d[3] = (uint32_t)(ga >> 32) | (2u << 30);     // global_addr[56:32], type=2
    v8i g1d;
    g1d[0] = (2 << 16) | (1 << 20) | (3 << 22);     // data_size=4B, pad_en, ivl=16dw, amt=1dw
    g1d[1] = (int)((L_ & 0xFFFF) << 16);            // tensor_dim0[15:0] = 8192
    g1d[2] = 0;                                     // tensor_dim0[31:16]=0, tensor_dim1[15:0]=0
    g1d[3] = 16 | (TT_ << 16);                      // tensor_dim1 hi, tile_dim0=256
    g1d[4] = TC_;                                   // tile_dim1=16, tile_dim2=0
    g1d[5] = L_;                                    // tensor_dim0_stride lo = 8192
    g1d[6] = 0;                                     // stride hi, dim1_stride lo
    g1d[7] = 0;
    v4i zz4 = {0, 0, 0, 0};
    v8i zz8 = {0, 0, 0, 0, 0, 0, 0, 0};
    __builtin_amdgcn_tensor_load_to_lds(g0d, g1d, zz4, zz4, zz8, 0);
  }

  // ---- stage zero-padded filter rows for the 4 groups of this tile ----
  for (int e = tid; e < 4 * 160; e += 256) {
    int gl  = e / 160;
    int tau = (e % 160) - 16;
    float kv = 0.0f;
    if (tau >= 0 && tau < LC_) kv = kbuf[(g0 + gl) * LC_ + tau];
    lds_k[e] = kv;
  }

  // ---- stage z = x2*v window [T0-128, T0+256), causal zero-fill for t<0 ----
#pragma unroll
  for (int it = 0; it < 6; ++it) {
    int widx = tid + it * 256;                 // [0, 16*96)
    int ch = widx / 96;
    int w0 = (widx % 96) * 4;
    long tg = (long)T0 - 128 + w0;
    float z0 = 0.f, z1 = 0.f, z2 = 0.f, z3 = 0.f;
    if (tg >= 0) {
      const float4 a  = *(const float4*)(x2 + (size_t)(cbase + ch) * L_ + tg);
      const float4 bb = *(const float4*)(v  + (size_t)(cbase + ch) * L_ + tg);
      z0 = a.x * bb.x; z1 = a.y * bb.y; z2 = a.z * bb.z; z3 = a.w * bb.w;
    }
    float* row = lds_z + ch * 408;
    row[skew(w0 + 0)] = z0; row[skew(w0 + 1)] = z1;
    row[skew(w0 + 2)] = z2; row[skew(w0 + 3)] = z3;
  }

  // warm L2 for the next time tile (global_prefetch_b8)
  if (T0 + TT_ + 256 <= L_) {
    __builtin_prefetch(x2 + (size_t)cbase * L_ + T0 + TT_ + (size_t)(tid & 31) * 16, 0, 1);
    __builtin_prefetch(v  + (size_t)cbase * L_ + T0 + TT_ + (size_t)(tid & 31) * 16, 0, 1);
  }

  // issuing wave drains the TDM transfer before the workgroup barrier
  if (tid < 32) {
    __builtin_amdgcn_s_wait_tensorcnt(0);
  }
  __syncthreads();

  // ---- WMMA FIR: each wave handles 2 same-group channels (shared A frag) ----
  const int wave = tid >> 5;
  const int lane = tid & 31;
  const int li   = lane & 15;                  // M (A) / N (B,C) index
  const int hi   = lane >> 4;                  // half-wave select
  const int ch0  = wave * 2;
  const int gl   = ch0 >> 2;

  const float bias0 = conv_bias[dbase + ch0];
  const float bias1 = conv_bias[dbase + ch0 + 1];

  v8f acc0 = {};
  v8f acc1 = {};
  const float* krow = lds_k + gl * 160;
  const float* zr0  = lds_z + ch0 * 408;
  const float* zr1  = zr0 + 408;

  // Y[i,n] = sum_c sum_q k_pad[i-q+c] * z[T0 + 16n + q - c], c = 0,16,...,128
  for (int cb = 0; cb <= 128; cb += 16) {
#pragma unroll
    for (int q0 = 0; q0 < 16; q0 += 4) {
      v2f a, b0, b1;
#pragma unroll
      for (int j = 0; j < 2; ++j) {
        // A 16x4 f32 layout: vgpr j holds K = q0+j (lanes 0-15), q0+j+2 (lanes 16-31)
        int q   = q0 + j + 2 * hi;
        int tau = li - q + cb;                 // in [-15, 143], zero-padded
        a[j] = krow[tau + 16];
        int w = 16 * li + q - cb + 128;        // in [0, 384)
        b0[j] = zr0[skew(w)];
        b1[j] = zr1[skew(w)];
      }
      acc0 = __builtin_amdgcn_wmma_f32_16x16x4_f32(false, a, false, b0,
                                                   (short)0, acc0, false, false);
      acc1 = __builtin_amdgcn_wmma_f32_16x16x4_f32(false, a, false, b1,
                                                   (short)0, acc1, false, false);
    }
  }

  // ---- epilogue: (y + z*bias)*x1, rotated write into transpose staging ----
  const float* x1r0 = lds_x1 + ch0 * 272;
  const float* x1r1 = x1r0 + 272;
#pragma unroll
  for (int r = 0; r < 8; ++r) {
    int i  = r + 8 * hi;                       // C/D layout: M = r (+8 for hi lanes)
    int tl = 16 * li + i;                      // local time in [0,256)
    float z0 = zr0[skew(tl + 128)];
    float z1 = zr1[skew(tl + 128)];
    float o0 = (acc0[r] + z0 * bias0) * x1r0[skew(tl)];
    float o1 = (acc1[r] + z1 * bias1) * x1r1[skew(tl)];
    lds_o[tl * TC_ + ((ch0 + tl) & 15)]     = o0;
    lds_o[tl * TC_ + ((ch0 + 1 + tl) & 15)] = o1;
  }

  __syncthreads();

  // ---- coalesced float4 stores of out[b, T0+t, dbase..dbase+15] ----
#pragma unroll
  for (int it = 0; it < 4; ++it) {
    int idx = tid + it * 256;                  // [0, 1024) float4 units
    int t = idx >> 2;
    int q = (idx & 3) * 4;
    float4 o;
    o.x = lds_o[t * TC_ + ((q + 0 + t) & 15)];
    o.y = lds_o[t * TC_ + ((q + 1 + t) & 15)];
    o.z = lds_o[t * TC_ + ((q + 2 + t) & 15)];
    o.w = lds_o[t * TC_ + ((q + 3 + t) & 15)];
    float* po = out + ((size_t)b * L_ + T0 + t) * D_ + dbase + q;
    *(float4*)po = o;
  }
}

// ---------------------------------------------------------------------------
extern "C" void kernel_launch(void* const* d_in, const int* in_sizes, int n_in,
                              void* d_out, int out_size, void* d_ws, size_t ws_size,
                              hipStream_t stream) {
  const float* x1 = (const float*)d_in[0];
  const float* x2 = (const float*)d_in[1];
  const float* v  = (const float*)d_in[2];
  const float* h  = (const float*)d_in[3];
  const float* cb = (const float*)d_in[4];
  float* out  = (float*)d_out;
  float* kbuf = (float*)d_ws;                  // G_*LC_*4 = 128 KB scratch

  hipLaunchKernelGGL(hyena_filter_precompute, dim3(G_), dim3(LC_), 0, stream, h, kbuf);

  dim3 grid(L_ / TT_, (B_ * D_) / TC_);        // 32 x 128 workgroups
  hipLaunchKernelGGL(hyena_conv_kernel, grid, dim3(256), 0, stream,
                     x1, x2, v, kbuf, cb, out);
}